// GCNLayer_37503654428949
// MI455X (gfx1250) — compile-verified
//
#include <hip/hip_runtime.h>
#include <hip/hip_bf16.h>

typedef __attribute__((ext_vector_type(2))) float v2f;
typedef __attribute__((ext_vector_type(8))) float v8f;

#define D 128           // D_IN == D_OUT == 128
#define TILES_N 8       // 128 / 16 output-column tiles
#define KSTEPS 32       // 128 / 4 K-steps for f32 WMMA

// ---------------------------------------------------------------------------
// Kernel 0: zero fill (float4 fast path + scalar tail)
// ---------------------------------------------------------------------------
__global__ void gcn_zero_kernel(float* __restrict__ p, int n) {
    int i = blockIdx.x * blockDim.x + threadIdx.x;
    int i4 = i * 4;
    if (i4 + 3 < n) {
        *(float4*)(p + i4) = make_float4(0.f, 0.f, 0.f, 0.f);
    } else {
        for (int j = i4; j < n; ++j) p[j] = 0.f;
    }
}

// ---------------------------------------------------------------------------
// Kernel 1: wave-per-edge scatter-add of raw features + in-degree counts.
// Each of the 32 lanes handles 4 consecutive floats (float4) of the 128-wide
// row: gather from x[src], global_atomic_add_f32 into agg[dst].
// ---------------------------------------------------------------------------
__global__ void gcn_scatter_kernel(const float* __restrict__ x,
                                   const int* __restrict__ edge_index,
                                   float* __restrict__ agg,
                                   float* __restrict__ cnt,
                                   int nEdges) {
    int gid  = blockIdx.x * blockDim.x + threadIdx.x;
    int wave = gid >> 5;
    int lane = threadIdx.x & 31;
    if (wave >= nEdges) return;

    int src = edge_index[wave];            // row 0 of [2, E]
    int dst = edge_index[nEdges + wave];   // row 1 of [2, E]

    const float4 v = *(const float4*)(x + (size_t)src * D + lane * 4);
    float* a = agg + (size_t)dst * D + lane * 4;
    atomicAdd(a + 0, v.x);
    atomicAdd(a + 1, v.y);
    atomicAdd(a + 2, v.z);
    atomicAdd(a + 3, v.w);
    if (lane == 0) atomicAdd(cnt + dst, 1.0f);
}

// ---------------------------------------------------------------------------
// Kernel 2: dense transform with fp32 WMMA, fused bias/mean/ReLU, IN PLACE.
// One wave owns 16 nodes. It preloads the full 16x128 A-tile into registers
// (so overwriting the same rows of `agg` afterwards is safe), then computes
// the 8 column tiles: C(16x16) += A(16x4) * B(4x16) over 32 K-steps.
//
// f32 WMMA A layout: lane L holds M = L%16, K = 2*(L/16) + {VGPR0,VGPR1}
// within each K-group of 4 -> per step s the lane loads float2 at col 4s+2h.
// B (= W^T slice, B[k][n] = W[n][k]) mirrors A with M->N, so lane L loads
// float2 from row W[t*16 + L%16] at col 4s+2h.
// C/D layout: VGPR r holds row M = r + 8*(lane>=16), column N = lane%16.
// ---------------------------------------------------------------------------
__global__ void gcn_wmma_kernel(float* __restrict__ agg,      // in: sums, out: result
                                const float* __restrict__ Wm, // [128,128], row o, col k
                                const float* __restrict__ bias,
                                const float* __restrict__ cnt,
                                int nNodes) {
    int gid    = blockIdx.x * blockDim.x + threadIdx.x;
    int wave   = gid >> 5;
    int lane   = threadIdx.x & 31;
    int m_base = wave * 16;
    if (m_base >= nNodes) return;

    const int mrow = lane & 15;
    const int h    = lane >> 4;

    // Preload full A tile (16x128) -> 64 VGPRs
    const float* arow = agg + (size_t)(m_base + mrow) * D + 2 * h;
    v2f a[KSTEPS];
#pragma unroll
    for (int s = 0; s < KSTEPS; ++s) a[s] = *(const v2f*)(arow + 4 * s);

    // Per-row counts for the epilogue (row M = r + 8*h)
    float cnt8[8];
#pragma unroll
    for (int r = 0; r < 8; ++r) cnt8[r] = cnt[m_base + r + 8 * h];

    for (int t = 0; t < TILES_N; ++t) {
        const float* wrow = Wm + (size_t)(t * 16 + mrow) * D + 2 * h;
        v8f c = {};
#pragma unroll
        for (int s = 0; s < KSTEPS; ++s) {
            v2f b = *(const v2f*)(wrow + 4 * s);
            // D = A(16x4,f32) * B(4x16,f32) + C(16x16,f32)
            c = __builtin_amdgcn_wmma_f32_16x16x4_f32(
                    /*neg_a=*/false, a[s], /*neg_b=*/false, b,
                    /*c_mod=*/(short)0, c, /*reuse_a=*/false, /*reuse_b=*/false);
        }

        const int   col = t * 16 + mrow;
        const float bv  = bias[col];
#pragma unroll
        for (int r = 0; r < 8; ++r) {
            int   node  = m_base + r + 8 * h;
            float cv    = cnt8[r];
            float denom = fmaxf(cv, 1.0f);
            float val   = (c[r] + cv * bv) / denom;  // mean with bias folded in
            val         = fmaxf(val, 0.0f);          // ReLU
            agg[(size_t)node * D + col] = val;
        }
    }
}

// ---------------------------------------------------------------------------
// Launcher
//   d_in[0] x              [100000,128] f32
//   d_in[1] edge_index     [2,1600000]  int
//   d_in[2] edge_features  [1600000]    f32 (unused by reference)
//   d_in[3] W              [128,128]    f32
//   d_in[4] b              [128]        f32
//   d_out  = [100000,128] f32 (also used as the aggregation buffer)
//   d_ws   : counts [100000] f32
// ---------------------------------------------------------------------------
extern "C" void kernel_launch(void* const* d_in, const int* in_sizes, int n_in,
                              void* d_out, int out_size, void* d_ws, size_t ws_size,
                              hipStream_t stream) {
    const float* x   = (const float*)d_in[0];
    const int*   ei  = (const int*)d_in[1];
    const float* Wm  = (const float*)d_in[3];
    const float* bia = (const float*)d_in[4];
    float*       out = (float*)d_out;
    float*       cnt = (float*)d_ws;

    const int nNodes = in_sizes[0] / D;      // 100000
    const int nEdges = in_sizes[1] / 2;      // 1600000

    // Phase 0: zero aggregation buffer (d_out) and counts
    {
        int n   = nNodes * D;
        int thr = (n + 3) / 4;
        gcn_zero_kernel<<<(thr + 255) / 256, 256, 0, stream>>>(out, n);
        int thr2 = (nNodes + 3) / 4;
        gcn_zero_kernel<<<(thr2 + 255) / 256, 256, 0, stream>>>(cnt, nNodes);
    }

    // Phase 1: scatter-add raw features + counts (wave per edge)
    {
        long long threads = (long long)nEdges * 32;
        int blocks = (int)((threads + 255) / 256);
        gcn_scatter_kernel<<<blocks, 256, 0, stream>>>(x, ei, out, cnt, nEdges);
    }

    // Phase 2: fp32 WMMA transform + bias/mean/ReLU, in place on d_out
    {
        int waves   = (nNodes + 15) / 16;
        long long threads = (long long)waves * 32;
        int blocks  = (int)((threads + 255) / 256);
        gcn_wmma_kernel<<<blocks, 256, 0, stream>>>(out, Wm, bia, cnt, nNodes);
    }
}